// GCN_17377437680138
// MI455X (gfx1250) — compile-verified
//
#include <hip/hip_runtime.h>
#include <hip/hip_bf16.h>

// ---------------------------------------------------------------------------
// GCN (3x GCNConv + ReLU + global_add_pool concat) for MI455X / gfx1250.
//   N = 100000 nodes (multiple of 16), E = 1.6M edges, F = H = 64, G = 64.
// Strategy:
//   * fp32 WMMA (V_WMMA_F32_16X16X4_F32) for the dense xw = h @ W GEMMs
//     (GEMM is <1% of runtime -> keep full fp32 accuracy, still on matrix pipe)
//   * edge scatter with non-returning global f32 atomics; working set (~77MB)
//     fits in the 192MB L2, so this runs at L2 atomic rate, not HBM.
//   * self-loop term folded into finalize (agg + xw*dinv^2 + b -> relu),
//     pooling fused there too. h written in-place over agg.
// ---------------------------------------------------------------------------

typedef __attribute__((ext_vector_type(2))) float v2f;
typedef __attribute__((ext_vector_type(8))) float v8f;

#define FEAT 64
#define POOL_W 192   // 3 * HIDDEN

__global__ __launch_bounds__(256) void gcn_zero_f32(float* __restrict__ p, int n) {
  int i = blockIdx.x * blockDim.x + threadIdx.x;
  if (i < n) p[i] = 0.0f;
}

// deg[col[e]] += 1  (edge targets only; self loop added in dinv kernel)
__global__ __launch_bounds__(256) void gcn_degree(const int* __restrict__ col,
                                                  float* __restrict__ deg, int nEdges) {
  int e = blockIdx.x * blockDim.x + threadIdx.x;
  if (e < nEdges)
    __hip_atomic_fetch_add(&deg[col[e]], 1.0f, __ATOMIC_RELAXED,
                           __HIP_MEMORY_SCOPE_AGENT);
}

// dinv[i] = rsqrt(deg[i] + 1)   (+1 = self loop; always >= 1, matches reference)
__global__ __launch_bounds__(256) void gcn_dinv(float* __restrict__ deg, int n) {
  int i = blockIdx.x * blockDim.x + threadIdx.x;
  if (i < n) deg[i] = rsqrtf(deg[i] + 1.0f);
}

// XW[nRows,64] = X[nRows,64] @ W[64,64] via V_WMMA_F32_16X16X4_F32.
// One 16x16 output tile per wave; nRows must be a multiple of 16 (100000 is).
__global__ __launch_bounds__(256) void gcn_gemm_wmma(const float* __restrict__ X,
                                                     const float* __restrict__ W,
                                                     float* __restrict__ XW,
                                                     int nRows) {
  const int lane = threadIdx.x & 31;
  const int wave = threadIdx.x >> 5;
  const int tile = blockIdx.x * 8 + wave;          // wave-uniform
  const int nTiles = (nRows >> 4) * 4;             // (rows/16) * (64/16)
  if (tile >= nTiles) return;                      // whole-wave exit: EXEC stays full
  const int tm   = tile >> 2;                      // row tile
  const int tn   = tile & 3;                       // col tile
  const int m    = lane & 15;
  const int half = lane >> 4;                      // 0 or 1

  v8f c = {};
  const float* __restrict__ xrow = X + (size_t)(tm * 16 + m) * FEAT;
  const float* __restrict__ wcol = W + tn * 16 + m;
#pragma unroll
  for (int k = 0; k < 16; ++k) {
    const int kk = k * 4 + half * 2;               // ISA 16x4 f32 A / 4x16 f32 B layout
    v2f a, b;
    a.x = xrow[kk];
    a.y = xrow[kk + 1];
    b.x = wcol[kk * FEAT];
    b.y = wcol[(kk + 1) * FEAT];
    c = __builtin_amdgcn_wmma_f32_16x16x4_f32(false, a, false, b,
                                              (short)0, c, false, false);
  }
  // D layout: vgpr i -> row tm*16 + 8*half + i, col tn*16 + m
  float* __restrict__ orow = XW + (size_t)(tm * 16 + half * 8) * FEAT + tn * 16 + m;
#pragma unroll
  for (int i = 0; i < 8; ++i) orow[(size_t)i * FEAT] = c[i];
}

// agg[col[e]] += XW[row[e]] * (dinv[row[e]] * dinv[col[e]]);  one wave per edge,
// 2 floats/lane (coalesced 256B gather), non-returning f32 atomics for scatter.
__global__ __launch_bounds__(256) void gcn_edge_scatter(const float* __restrict__ XW,
                                                        const int* __restrict__ row,
                                                        const int* __restrict__ col,
                                                        const float* __restrict__ dinv,
                                                        float* __restrict__ agg,
                                                        int nEdges) {
  const int lane   = threadIdx.x & 31;
  const int wave   = (int)((blockIdx.x * blockDim.x + threadIdx.x) >> 5);
  const int nWaves = (int)((gridDim.x * blockDim.x) >> 5);
  for (int e = wave; e < nEdges; e += nWaves) {
    const int en = e + nWaves;                     // prefetch next source row
    if (en < nEdges)
      __builtin_prefetch(XW + (size_t)row[en] * FEAT + lane * 2, 0, 0);

    const int   r = row[e];
    const int   t = col[e];
    const float w = dinv[r] * dinv[t];
    const float2 v = *(const float2*)(XW + (size_t)r * FEAT + lane * 2);
    float* dst = agg + (size_t)t * FEAT + lane * 2;
    __hip_atomic_fetch_add(dst,     v.x * w, __ATOMIC_RELAXED, __HIP_MEMORY_SCOPE_AGENT);
    __hip_atomic_fetch_add(dst + 1, v.y * w, __ATOMIC_RELAXED, __HIP_MEMORY_SCOPE_AGENT);
  }
}

// h = relu(agg + xw*dinv^2 + bias); write h in-place into agg; pool into d_out.
__global__ __launch_bounds__(256) void gcn_finalize(float* __restrict__ agg,
                                                    const float* __restrict__ XW,
                                                    const float* __restrict__ dinv,
                                                    const float* __restrict__ bias,
                                                    const int* __restrict__ batch,
                                                    float* __restrict__ pooled,
                                                    int layerOff, int nNodes) {
  int idx = blockIdx.x * blockDim.x + threadIdx.x;
  if (idx >= nNodes * FEAT) return;
  const int node = idx >> 6;
  const int f    = idx & 63;
  const float di = dinv[node];
  float h = agg[idx] + XW[idx] * di * di + bias[f];
  h = fmaxf(h, 0.0f);
  agg[idx] = h;
  __hip_atomic_fetch_add(&pooled[(size_t)batch[node] * POOL_W + layerOff + f], h,
                         __ATOMIC_RELAXED, __HIP_MEMORY_SCOPE_AGENT);
}

extern "C" void kernel_launch(void* const* d_in, const int* in_sizes, int n_in,
                              void* d_out, int out_size, void* d_ws, size_t ws_size,
                              hipStream_t stream) {
  const float* x     = (const float*)d_in[0];
  const int*   ei    = (const int*)d_in[1];
  const int*   batch = (const int*)d_in[2];
  const float* Ws[3] = {(const float*)d_in[3], (const float*)d_in[5], (const float*)d_in[7]};
  const float* bs[3] = {(const float*)d_in[4], (const float*)d_in[6], (const float*)d_in[8]};

  const int N = in_sizes[0] / FEAT;   // 100000 nodes
  const int E = in_sizes[1] / 2;      // 1600000 edges
  const int* rowI = ei;               // edge_index[0] = sources
  const int* colI = ei + E;           // edge_index[1] = targets
  float* out = (float*)d_out;

  // workspace carve-up (offsets in floats, 4KB-aligned slots)
  float* ws = (float*)d_ws;
  const size_t degSz = ((size_t)N + 1023) & ~(size_t)1023;
  const size_t matSz = (size_t)N * FEAT;
  float* deg  = ws;                   // becomes dinv in place
  float* xw   = ws + degSz;
  float* aggA = xw + matSz;
  float* aggB = aggA + matSz;

  const int T256_out = (out_size + 255) / 256;
  const int T256_N   = (N + 255) / 256;
  const int T256_E   = (E + 255) / 256;
  const int T256_M   = (int)((matSz + 255) / 256);

  gcn_zero_f32<<<T256_out, 256, 0, stream>>>(out, out_size);
  gcn_zero_f32<<<T256_N, 256, 0, stream>>>(deg, N);
  gcn_degree<<<T256_E, 256, 0, stream>>>(colI, deg, E);
  gcn_dinv<<<T256_N, 256, 0, stream>>>(deg, N);

  const int gemmBlocks = ((N >> 4) * 4 + 7) / 8;   // 8 waves (tiles) per block

  const float* hin = x;
  float* aggs[3] = {aggA, aggB, aggA};             // h1 dead after layer-2 GEMM
  for (int l = 0; l < 3; ++l) {
    float* agg = aggs[l];
    gcn_zero_f32<<<T256_M, 256, 0, stream>>>(agg, (int)matSz);
    gcn_gemm_wmma<<<gemmBlocks, 256, 0, stream>>>(hin, Ws[l], xw, N);
    gcn_edge_scatter<<<8192, 256, 0, stream>>>(xw, rowI, colI, deg, agg, E);
    gcn_finalize<<<T256_M, 256, 0, stream>>>(agg, xw, deg, bs[l], batch, out,
                                             l * FEAT, N);
    hin = agg;                                     // h for next layer (in-place)
  }
}